// GATLayer_25898652795467
// MI455X (gfx1250) — compile-verified
//
#include <hip/hip_runtime.h>
#include <hip/hip_bf16.h>
#include <math.h>

#define N_NODES 40000
#define N_EDGES 640000
#define IN_F    128
#define NHEAD   8
#define FPH     16
#define E_TOT   (N_EDGES + N_NODES)   // 680000 edges incl. self-loops
#define NEG_SLOPE 0.2f
#define BN_EPS    1e-5f
#define SM_EPS    1e-16f

typedef float v2f __attribute__((ext_vector_type(2)));
typedef float v8f __attribute__((ext_vector_type(8)));

// ---------------------------------------------------------------------------
// fp32 atomic max via monotone int/uint reinterpretation (lowers to
// global_atomic_max_i32 / global_atomic_min_u32, no CAS loop)
__device__ __forceinline__ void atomicMaxF32(float* addr, float val) {
    if (val >= 0.0f) {
        atomicMax((int*)addr, __float_as_int(val));
    } else {
        atomicMin((unsigned int*)addr, (unsigned int)__float_as_int(val));
    }
}

// ---------------------------------------------------------------------------
// Kernel 1: xp = x @ W via V_WMMA_F32_16X16X4_F32.
// Block = 256 threads = 8 waves; block b owns rows [16b,16b+16); wave w owns
// cols [16w,16w+16). K=128 -> 32 chained K=4 WMMAs.
__global__ __launch_bounds__(256) void gemm_xp_wmma(
        const float* __restrict__ x, const float* __restrict__ W,
        float* __restrict__ xp) {
    const int wave = threadIdx.x >> 5;
    const int lane = threadIdx.x & 31;
    const int rowBase = blockIdx.x * 16;
    const int colBase = wave * 16;
    const int m    = lane & 15;        // row (A) / col (B) within tile
    const int half = lane >> 4;        // 0: K pair {0,1}, 1: K pair {2,3}
    const int kk   = half * 2;

    const float* __restrict__ xr = x + (size_t)(rowBase + m) * IN_F;

    v8f acc = {};
    for (int k0 = 0; k0 < IN_F; k0 += 4) {
        v2f a, b;
        a.x = xr[k0 + kk];
        a.y = xr[k0 + kk + 1];
        b.x = W[(size_t)(k0 + kk)     * IN_F + colBase + m];
        b.y = W[(size_t)(k0 + kk + 1) * IN_F + colBase + m];
        acc = __builtin_amdgcn_wmma_f32_16x16x4_f32(
                false, a, false, b, (short)0, acc, false, false);
    }

    // C/D layout: VGPR r -> row (r + 8*half), col = lane&15
    const int col = colBase + m;
    const int rb  = rowBase + half * 8;
#pragma unroll
    for (int r = 0; r < 8; ++r) {
        xp[(size_t)(rb + r) * IN_F + col] = acc[r];
    }
}

// ---------------------------------------------------------------------------
// Kernel 2: per-(node,head) attention scores a_s, a_d
__global__ void attn_scores(const float* __restrict__ xp,
                            const float* __restrict__ att_src,
                            const float* __restrict__ att_dst,
                            float* __restrict__ a_s, float* __restrict__ a_d) {
    int idx = blockIdx.x * blockDim.x + threadIdx.x;
    if (idx >= N_NODES * NHEAD) return;
    int n = idx >> 3;
    int h = idx & 7;
    const float* v  = xp + (size_t)n * IN_F + h * FPH;
    const float* as = att_src + h * FPH;
    const float* ad = att_dst + h * FPH;
    float ss = 0.f, sd = 0.f;
#pragma unroll
    for (int f = 0; f < FPH; ++f) {
        float xv = v[f];
        ss += xv * as[f];
        sd += xv * ad[f];
    }
    a_s[idx] = ss;
    a_d[idx] = sd;
}

// ---------------------------------------------------------------------------
// generic float fill (workspace / output init each call -> graph-replay safe)
__global__ void fill_f32(float* __restrict__ p, float v, int n) {
    int i = blockIdx.x * blockDim.x + threadIdx.x;
    if (i < n) p[i] = v;
}

// ---------------------------------------------------------------------------
// Kernel 3: per-(edge,head) leaky-relu logit + segment max into m[dst,h]
__global__ void edge_logit_max(const int* __restrict__ ei,
                               const float* __restrict__ a_s,
                               const float* __restrict__ a_d,
                               float* __restrict__ e_buf,
                               float* __restrict__ m_buf) {
    int idx = blockIdx.x * blockDim.x + threadIdx.x;
    if (idx >= E_TOT * NHEAD) return;
    int e = idx >> 3;
    int h = idx & 7;
    int src, dst;
    if (e < N_EDGES) { src = ei[e]; dst = ei[N_EDGES + e]; }
    else             { src = dst = e - N_EDGES; }           // self-loop
    float v = a_s[src * NHEAD + h] + a_d[dst * NHEAD + h];
    v = (v > 0.f) ? v : NEG_SLOPE * v;
    e_buf[idx] = v;
    atomicMaxF32(&m_buf[dst * NHEAD + h], v);
}

// ---------------------------------------------------------------------------
// Kernel 4: p = exp(e - m[dst]); segment sum into s[dst,h]; e_buf := p
__global__ void edge_exp_sum(const int* __restrict__ ei,
                             const float* __restrict__ m_buf,
                             float* __restrict__ e_buf,
                             float* __restrict__ s_buf) {
    int idx = blockIdx.x * blockDim.x + threadIdx.x;
    if (idx >= E_TOT * NHEAD) return;
    int e = idx >> 3;
    int h = idx & 7;
    int dst = (e < N_EDGES) ? ei[N_EDGES + e] : (e - N_EDGES);
    float p = __expf(e_buf[idx] - m_buf[dst * NHEAD + h]);
    e_buf[idx] = p;
    atomicAdd(&s_buf[dst * NHEAD + h], p);
}

// ---------------------------------------------------------------------------
// Kernel 5: scatter messages: out[dst,c] += (p/(s+eps)) * xp[src,c]
// one thread per (edge, channel); 87M f32 atomic adds resolved in L2
__global__ void edge_scatter(const int* __restrict__ ei,
                             const float* __restrict__ p_buf,
                             const float* __restrict__ s_buf,
                             const float* __restrict__ xp,
                             float* __restrict__ out) {
    int idx = blockIdx.x * blockDim.x + threadIdx.x;
    if (idx >= E_TOT * IN_F) return;
    int e = idx >> 7;
    int c = idx & 127;
    int h = c >> 4;
    int src, dst;
    if (e < N_EDGES) { src = ei[e]; dst = ei[N_EDGES + e]; }
    else             { src = dst = e - N_EDGES; }
    float alpha = p_buf[e * NHEAD + h] / (s_buf[dst * NHEAD + h] + SM_EPS);
    atomicAdd(&out[(size_t)dst * IN_F + c],
              alpha * xp[(size_t)src * IN_F + c]);
}

// ---------------------------------------------------------------------------
// Kernel 6: per-channel sum / sumsq of (out + bias) for BatchNorm stats
__global__ __launch_bounds__(128) void bn_stats(const float* __restrict__ out,
                                                const float* __restrict__ bias,
                                                float* __restrict__ sums) {
    int c = threadIdx.x;            // channel 0..127
    float b = bias[c];
    float s = 0.f, s2 = 0.f;
    for (int r = blockIdx.x; r < N_NODES; r += gridDim.x) {
        float v = out[(size_t)r * IN_F + c] + b;
        s  += v;
        s2 += v * v;
    }
    atomicAdd(&sums[c], s);
    atomicAdd(&sums[IN_F + c], s2);
}

// ---------------------------------------------------------------------------
// Kernel 7: fused BN (batch stats) + ReLU + residual, in-place on d_out
__global__ void bn_finalize(float* __restrict__ out,
                            const float* __restrict__ x,
                            const float* __restrict__ bias,
                            const float* __restrict__ gamma,
                            const float* __restrict__ beta,
                            const float* __restrict__ sums) {
    int idx = blockIdx.x * blockDim.x + threadIdx.x;
    if (idx >= N_NODES * IN_F) return;
    int c = idx & 127;
    const float invN = 1.0f / (float)N_NODES;
    float mean = sums[c] * invN;
    float var  = sums[IN_F + c] * invN - mean * mean;   // biased variance
    float h = (out[idx] + bias[c] - mean) * rsqrtf(var + BN_EPS) * gamma[c]
              + beta[c];
    h = fmaxf(h, 0.f);
    out[idx] = h + x[idx];
}

// ---------------------------------------------------------------------------
extern "C" void kernel_launch(void* const* d_in, const int* in_sizes, int n_in,
                              void* d_out, int out_size, void* d_ws, size_t ws_size,
                              hipStream_t stream) {
    const float* x        = (const float*)d_in[0];
    const int*   ei       = (const int*)  d_in[1];   // [2,E] flat: src then dst
    const float* W        = (const float*)d_in[2];
    const float* att_src  = (const float*)d_in[3];
    const float* att_dst  = (const float*)d_in[4];
    const float* bias     = (const float*)d_in[5];
    const float* bn_gamma = (const float*)d_in[6];
    const float* bn_beta  = (const float*)d_in[7];
    float* out = (float*)d_out;                      // also the aggregation accumulator

    // workspace layout (floats)
    float* ws   = (float*)d_ws;
    float* xp   = ws;                                   // N*128   = 5,120,000
    float* a_s  = xp  + (size_t)N_NODES * IN_F;         // N*8
    float* a_d  = a_s + (size_t)N_NODES * NHEAD;        // N*8
    float* mbuf = a_d + (size_t)N_NODES * NHEAD;        // N*8
    float* sbuf = mbuf + (size_t)N_NODES * NHEAD;       // N*8
    float* ebuf = sbuf + (size_t)N_NODES * NHEAD;       // E_TOT*8 = 5,440,000
    float* sums = ebuf + (size_t)E_TOT * NHEAD;         // 256

    // --- init (required every call: harness poisons but never re-poisons) ---
    fill_f32<<<(N_NODES * IN_F + 255) / 256, 256, 0, stream>>>(out, 0.f, N_NODES * IN_F);
    fill_f32<<<(N_NODES * NHEAD + 255) / 256, 256, 0, stream>>>(sbuf, 0.f, N_NODES * NHEAD);
    fill_f32<<<(N_NODES * NHEAD + 255) / 256, 256, 0, stream>>>(mbuf, -INFINITY, N_NODES * NHEAD);
    fill_f32<<<1, 256, 0, stream>>>(sums, 0.f, 2 * IN_F);

    // --- 1) xp = x @ W (WMMA fp32) ---
    gemm_xp_wmma<<<N_NODES / 16, 256, 0, stream>>>(x, W, xp);

    // --- 2) attention scores ---
    attn_scores<<<(N_NODES * NHEAD + 255) / 256, 256, 0, stream>>>(
        xp, att_src, att_dst, a_s, a_d);

    // --- 3) logits + segment max ---
    edge_logit_max<<<(E_TOT * NHEAD + 255) / 256, 256, 0, stream>>>(
        ei, a_s, a_d, ebuf, mbuf);

    // --- 4) exp + segment sum ---
    edge_exp_sum<<<(E_TOT * NHEAD + 255) / 256, 256, 0, stream>>>(
        ei, mbuf, ebuf, sbuf);

    // --- 5) scatter-aggregate messages into out ---
    edge_scatter<<<(E_TOT * IN_F + 255) / 256, 256, 0, stream>>>(
        ei, ebuf, sbuf, xp, out);

    // --- 6) BN statistics ---
    bn_stats<<<256, 128, 0, stream>>>(out, bias, sums);

    // --- 7) BN + ReLU + residual (in place) ---
    bn_finalize<<<(N_NODES * IN_F + 255) / 256, 256, 0, stream>>>(
        out, x, bias, bn_gamma, bn_beta, sums);
}